// SegmentationModel_2894807957809
// MI455X (gfx1250) — compile-verified
//
#include <hip/hip_runtime.h>
#include <hip/hip_bf16.h>
#include <math.h>

// ---------------------------------------------------------------------------
// DGCNN part-segmentation forward pass for MI455X (gfx1250, wave32, WMMA).
//
// All heavy math (kNN inner products + 1x1-conv GEMMs) runs on
// V_WMMA_F32_16X16X4_F32. Channel count C is a template parameter so the
// K-loops unroll, guards constant-fold, and loads batch into clauses with
// immediate offsets. Distances never touch HBM: 16x128 strips are produced
// via WMMA into LDS and top-(K+1) selection happens there. BatchNorm batch
// statistics are fused into the GEMM epilogue (LDS ds_add_f32 + one global
// atomic per channel); normalization + LeakyReLU are applied lazily by the
// next consumer (from LDS-staged scale/shift).
// ---------------------------------------------------------------------------

#define BB   2
#define NP   10000
#define KK   10
#define KP1  11          // K+1 neighbors kept (first = self, dropped)
#define EPSI 1e-5f
#define NEG_SLOPE 0.2f

typedef float v2f __attribute__((ext_vector_type(2)));
typedef float v8f __attribute__((ext_vector_type(8)));

__device__ __forceinline__ float lrelu(float v) { return v >= 0.f ? v : NEG_SLOPE * v; }

__device__ __forceinline__ v8f wmma_f32(v2f a, v2f b, v8f c) {
  // D = A(16x4) * B(4x16) + C(16x16), fp32.
  return __builtin_amdgcn_wmma_f32_16x16x4_f32(false, a, false, b, (short)0, c,
                                               false, false);
}

// Channel read with compile-time tail guard (only C%4!=0 keeps the compare).
template <int C>
__device__ __forceinline__ float chan_val(const float* __restrict__ base, int c) {
  if (c < C) return base[(size_t)c * NP];
  return 0.f;
}

// Edge-feature read: c<C0 -> neighbor-center diff, c<C -> center, else 0.
template <int C, int C0>
__device__ __forceinline__ float edge_val(const float* __restrict__ pj,
                                          const float* __restrict__ pn, int c) {
  if (c < C0) return pj[(size_t)c * NP] - pn[(size_t)c * NP];
  else if (c < C) return pn[(size_t)(c - C0) * NP];
  return 0.f;
}

// ---------------------------------------------------------------------------
// Squared norms per point: xx[b][n] = sum_c x[b][c][n]^2
// ---------------------------------------------------------------------------
__global__ void sqnorm_kernel(const float* __restrict__ x, float* __restrict__ xx,
                              int C) {
  const int N = NP;
  int i = blockIdx.x * blockDim.x + threadIdx.x;
  if (i >= BB * N) return;
  int b = i / N, n = i % N;
  const float* xb = x + (size_t)b * C * N + n;
  float s = 0.f;
  for (int c = 0; c < C; ++c) { float v = xb[(size_t)c * N]; s += v * v; }
  xx[i] = s;
}

// ---------------------------------------------------------------------------
// kNN: per block, 16 query rows. A operand (query rows) hoisted to registers
// once; 8 waves compute 16x16 inner-product tiles per 128-column strip with
// fp32 WMMA; distances go to LDS; 16 threads/row keep private sorted top-11
// lists over column slices; merged per row at the end.
// Output: idxOut[b][n][k], k in [0,K), self excluded.
// ---------------------------------------------------------------------------
template <int C>
__global__ void __launch_bounds__(256) knn_kernel(const float* __restrict__ x,
                                                  const float* __restrict__ xx,
                                                  int* __restrict__ idxOut) {
  const int N = NP;
  constexpr int KSTEPS = (C + 3) / 4;
  int b  = blockIdx.y;
  int r0 = blockIdx.x * 16;
  const float* xb  = x  + (size_t)b * C * N;
  const float* xxb = xx + (size_t)b * N;

  int tid  = threadIdx.x;
  int wave = tid >> 5;
  int lane = tid & 31;
  int lo = lane & 15, hi = lane >> 4;
  int row = tid >> 4, slot = tid & 15;

  __shared__ float sdist[16 * 128];
  __shared__ float scd[16 * 16 * KP1];
  __shared__ int   sci[16 * 16 * KP1];

  float bd[KP1]; int bi[KP1];
#pragma unroll
  for (int q = 0; q < KP1; ++q) { bd[q] = INFINITY; bi[q] = 0; }

  float xr[8];
#pragma unroll
  for (int r = 0; r < 8; ++r) xr[r] = xxb[r0 + r + 8 * hi];

  // Hoisted A operand: rows r0+lo, channels ks*4 + 2*hi + {0,1}.
  const float* arow = xb + r0 + lo;
  v2f areg[KSTEPS];
#pragma unroll
  for (int ks = 0; ks < KSTEPS; ++ks) {
    int c0 = ks * 4 + 2 * hi;
    areg[ks].x = chan_val<C>(arow, c0);
    areg[ks].y = chan_val<C>(arow, c0 + 1);
  }

  const int nstrips = (N + 127) / 128;
  for (int s = 0; s < nstrips; ++s) {
    int cb0  = s * 128;
    int col  = cb0 + wave * 16 + lo;
    int colc = col < N ? col : N - 1;
    const float* bcol = xb + colc;
    if (col + 128 < N) __builtin_prefetch(xb + col + 128, 0, 0);

    v8f acc = {0.f, 0.f, 0.f, 0.f, 0.f, 0.f, 0.f, 0.f};
#pragma unroll
    for (int ks = 0; ks < KSTEPS; ++ks) {
      int c0 = ks * 4 + 2 * hi;
      v2f bv;
      bv.x = chan_val<C>(bcol, c0);
      bv.y = chan_val<C>(bcol, c0 + 1);
      acc = wmma_f32(areg[ks], bv, acc);
    }

    float xcn = xxb[colc];
#pragma unroll
    for (int r = 0; r < 8; ++r) {
      int m = r + 8 * hi;
      float d = xr[r] - 2.f * acc[r] + xcn;
      if (col >= N) d = INFINITY;
      sdist[m * 128 + wave * 16 + lo] = d;
    }
    __syncthreads();

    for (int cc = slot; cc < 128; cc += 16) {
      int g = cb0 + cc;
      float d = sdist[row * 128 + cc];
      if (g < N && d < bd[KP1 - 1]) {
        bd[KP1 - 1] = d; bi[KP1 - 1] = g;
#pragma unroll
        for (int q = KP1 - 1; q > 0; --q) {
          if (bd[q] < bd[q - 1]) {
            float td = bd[q]; bd[q] = bd[q - 1]; bd[q - 1] = td;
            int   ti = bi[q]; bi[q] = bi[q - 1]; bi[q - 1] = ti;
          }
        }
      }
    }
    __syncthreads();
  }

  // merge the 16 partial lists per row
#pragma unroll
  for (int q = 0; q < KP1; ++q) {
    scd[row * (16 * KP1) + slot * KP1 + q] = bd[q];
    sci[row * (16 * KP1) + slot * KP1 + q] = bi[q];
  }
  __syncthreads();
  if (slot == 0) {
    int base = row * (16 * KP1);
    for (int sel = 0; sel < KP1; ++sel) {
      float best = INFINITY; int bq = 0;
      for (int q = 0; q < 16 * KP1; ++q) {
        float v = scd[base + q];
        if (v < best) { best = v; bq = q; }
      }
      int gi = sci[base + bq];
      scd[base + bq] = INFINITY;
      if (sel > 0)
        idxOut[(size_t)b * N * KK + (size_t)(r0 + row) * KK + (sel - 1)] = gi;
    }
  }
}

// ---------------------------------------------------------------------------
// Generic fp32 WMMA GEMM: Y[b][m][p] = sum_c W[m][c] * X(c,p)   (raw output)
//   EDGE=false: X(c,p) = Xg[b][c][p], optionally lrelu(x*scale[c]+shift[c])
//   EDGE=true : p=(n,k); c<C0: xsrc[b][c][idx]-xsrc[b][c][n]; else center
// Each wave computes two 16x16 tiles (shared A = W) -> 16(M) x 128(P)/block.
// Epilogue: store raw Y, accumulate per-channel sum/sumsq (LDS -> global).
// Invalid (OOB) columns use clamped addresses; WMMA column independence
// means they cannot contaminate valid outputs, so no zeroing needed.
// ---------------------------------------------------------------------------
template <int C, int C0, bool EDGE, bool SCALED>
__global__ void __launch_bounds__(128) gemm_wmma_kernel(
    const float* __restrict__ Wm, int M,
    const float* __restrict__ Xg, int P, long ldXb,
    const float* __restrict__ inScale, const float* __restrict__ inShift,
    const float* __restrict__ xsrc, const int* __restrict__ idx,
    float* __restrict__ Y, float* __restrict__ gsum, float* __restrict__ gsq) {
  const int N = NP;
  constexpr int KSTEPS = (C + 3) / 4;
  int tid  = threadIdx.x;
  int wave = tid >> 5, lane = tid & 31;
  int lo = lane & 15, hi = lane >> 4;
  int b  = blockIdx.z;
  int m0 = blockIdx.y * 16;
  int p0 = blockIdx.x * 128 + wave * 32 + lo;
  int p1 = p0 + 16;
  bool pv0 = p0 < P, pv1 = p1 < P;
  int pc0 = pv0 ? p0 : P - 1;
  int pc1 = pv1 ? p1 : P - 1;

  __shared__ float s_sc[SCALED ? C : 1];
  __shared__ float s_sh[SCALED ? C : 1];
  if constexpr (SCALED) {
    for (int c = tid; c < C; c += 128) { s_sc[c] = inScale[c]; s_sh[c] = inShift[c]; }
  }
  __shared__ float s_sum[16], s_sq[16];
  if (tid < 16) { s_sum[tid] = 0.f; s_sq[tid] = 0.f; }
  __syncthreads();

  const float* pj0 = nullptr; const float* pn0 = nullptr;
  const float* pj1 = nullptr; const float* pn1 = nullptr;
  const float* bp0 = nullptr; const float* bp1 = nullptr;
  const size_t Pl = (size_t)P;
  if constexpr (EDGE) {
    const float* xsb = xsrc + (size_t)b * C0 * N;
    const int* idxb  = idx + (size_t)b * N * KK;
    int nn0 = pc0 / KK, nn1 = pc1 / KK;
    int jj0 = idxb[pc0], jj1 = idxb[pc1];
    pj0 = xsb + jj0; pn0 = xsb + nn0;
    pj1 = xsb + jj1; pn1 = xsb + nn1;
  } else {
    const float* Xb = Xg + (size_t)b * ldXb;
    bp0 = Xb + pc0 + (size_t)(2 * hi) * Pl;
    bp1 = Xb + pc1 + (size_t)(2 * hi) * Pl;
  }
  const float* Wr = Wm + (size_t)(m0 + lo) * C + 2 * hi;

  v8f acc0 = {0.f, 0.f, 0.f, 0.f, 0.f, 0.f, 0.f, 0.f};
  v8f acc1 = {0.f, 0.f, 0.f, 0.f, 0.f, 0.f, 0.f, 0.f};
#pragma unroll 8
  for (int ks = 0; ks < KSTEPS; ++ks) {
    int c0 = ks * 4 + 2 * hi;
    v2f wv;
    if constexpr (C % 4 == 0) {
      wv = *(const v2f*)(Wr + ks * 4);       // 8B-aligned pair load
    } else {
      wv.x = (c0     < C) ? Wr[ks * 4]     : 0.f;
      wv.y = (c0 + 1 < C) ? Wr[ks * 4 + 1] : 0.f;
    }
    v2f b0, b1;
    if constexpr (EDGE) {
      b0.x = edge_val<C, C0>(pj0, pn0, c0);
      b0.y = edge_val<C, C0>(pj0, pn0, c0 + 1);
      b1.x = edge_val<C, C0>(pj1, pn1, c0);
      b1.y = edge_val<C, C0>(pj1, pn1, c0 + 1);
    } else {
      size_t off = (size_t)ks * 4 * Pl;
      b0.x = bp0[off];      b0.y = bp0[off + Pl];
      b1.x = bp1[off];      b1.y = bp1[off + Pl];
      if constexpr (SCALED) {
        float s0 = s_sc[c0], h0 = s_sh[c0];
        float s1 = s_sc[c0 + 1], h1 = s_sh[c0 + 1];
        b0.x = lrelu(b0.x * s0 + h0);  b0.y = lrelu(b0.y * s1 + h1);
        b1.x = lrelu(b1.x * s0 + h0);  b1.y = lrelu(b1.y * s1 + h1);
      }
    }
    acc0 = wmma_f32(wv, b0, acc0);
    acc1 = wmma_f32(wv, b1, acc1);
  }

  float* Yb = Y + (size_t)b * M * P;
#pragma unroll
  for (int r = 0; r < 8; ++r) {
    int m = m0 + r + 8 * hi;
    float v0 = acc0[r], v1 = acc1[r];
    if (pv0) {
      Yb[(size_t)m * P + p0] = v0;
      atomicAdd(&s_sum[r + 8 * hi], v0);
      atomicAdd(&s_sq[r + 8 * hi], v0 * v0);
    }
    if (pv1) {
      Yb[(size_t)m * P + p1] = v1;
      atomicAdd(&s_sum[r + 8 * hi], v1);
      atomicAdd(&s_sq[r + 8 * hi], v1 * v1);
    }
  }
  __syncthreads();
  if (tid < 16) {
    atomicAdd(&gsum[m0 + tid], s_sum[tid]);
    atomicAdd(&gsq[m0 + tid], s_sq[tid]);
  }
}

// ---------------------------------------------------------------------------
// BN finalize: (sum,sumsq,gamma,beta) -> per-channel (scale, shift)
// ---------------------------------------------------------------------------
__global__ void finalize_bn_kernel(const float* __restrict__ gsum,
                                   const float* __restrict__ gsq,
                                   const float* __restrict__ g,
                                   const float* __restrict__ bb, int M,
                                   float invCnt, float* __restrict__ scale,
                                   float* __restrict__ shift) {
  int m = blockIdx.x * blockDim.x + threadIdx.x;
  if (m >= M) return;
  float mean = gsum[m] * invCnt;
  float var  = gsq[m] * invCnt - mean * mean;
  var = var > 0.f ? var : 0.f;
  float s = g[m] * rsqrtf(var + EPSI);
  scale[m] = s;
  shift[m] = bb[m] - mean * s;
}

// ---------------------------------------------------------------------------
// Max over K of lrelu(scale*y+shift):  x[b][c][n] = max_k ...
// ---------------------------------------------------------------------------
__global__ void maxpool_k_kernel(const float* __restrict__ Y,
                                 const float* __restrict__ sc,
                                 const float* __restrict__ sh,
                                 float* __restrict__ xout, int M) {
  const int N = NP;
  long i = (long)blockIdx.x * blockDim.x + threadIdx.x;
  long total = (long)BB * M * N;
  if (i >= total) return;
  int n = (int)(i % N);
  long t = i / N;
  int c = (int)(t % M);
  int b = (int)(t / M);
  const float* yb =
      Y + (size_t)b * M * N * KK + (size_t)c * N * KK + (size_t)n * KK;
  float s = sc[c], h = sh[c];
  float mx = -INFINITY;
  for (int k = 0; k < KK; ++k) mx = fmaxf(mx, lrelu(yb[k] * s + h));
  xout[(size_t)b * M * N + (size_t)c * N + n] = mx;
}

// ---------------------------------------------------------------------------
// Global max over N of lrelu(scale*y+shift): gfeat[b][m]
// ---------------------------------------------------------------------------
__global__ void __launch_bounds__(256) gmax_kernel(
    const float* __restrict__ Y, const float* __restrict__ sc,
    const float* __restrict__ sh, float* __restrict__ gfeat, int M) {
  const int N = NP;
  int m = blockIdx.x, b = blockIdx.y;
  const float* yb = Y + (size_t)b * M * N + (size_t)m * N;
  float s = sc[m], h = sh[m];
  float mx = -INFINITY;
  for (int n = threadIdx.x; n < N; n += 256)
    mx = fmaxf(mx, lrelu(yb[n] * s + h));
  __shared__ float red[256];
  red[threadIdx.x] = mx;
  __syncthreads();
  for (int o = 128; o > 0; o >>= 1) {
    if (threadIdx.x < o)
      red[threadIdx.x] = fmaxf(red[threadIdx.x], red[threadIdx.x + o]);
    __syncthreads();
  }
  if (threadIdx.x == 0) gfeat[(size_t)b * M + m] = red[0];
}

// ---------------------------------------------------------------------------
// Label branch: lfeat[b][m] = lrelu(bn(W6 @ label))  (stats over B only)
// ---------------------------------------------------------------------------
__global__ void label_kernel(const float* __restrict__ W6,
                             const float* __restrict__ label,
                             const float* __restrict__ g6,
                             const float* __restrict__ b6,
                             float* __restrict__ lfeat) {
  int m = threadIdx.x;
  if (m >= 64) return;
  float t[BB];
  for (int b = 0; b < BB; ++b) {
    float a = 0.f;
    for (int c = 0; c < 16; ++c) a += W6[m * 16 + c] * label[b * 16 + c];
    t[b] = a;
  }
  float mean = 0.f;
  for (int b = 0; b < BB; ++b) mean += t[b];
  mean /= (float)BB;
  float var = 0.f;
  for (int b = 0; b < BB; ++b) { float d = t[b] - mean; var += d * d; }
  var /= (float)BB;
  float s = g6[m] * rsqrtf(var + EPSI);
  float h = b6[m] - mean * s;
  for (int b = 0; b < BB; ++b) lfeat[b * 64 + m] = lrelu(t[b] * s + h);
}

// ---------------------------------------------------------------------------
// xc = concat(x1,x2,x3) along channels -> (B,192,N)
// ---------------------------------------------------------------------------
__global__ void concat3_kernel(const float* __restrict__ x1,
                               const float* __restrict__ x2,
                               const float* __restrict__ x3,
                               float* __restrict__ xc) {
  const int N = NP;
  long i = (long)blockIdx.x * blockDim.x + threadIdx.x;
  long total = (long)BB * 192 * N;
  if (i >= total) return;
  int n = (int)(i % N);
  long t = i / N;
  int c = (int)(t % 192);
  int b = (int)(t / 192);
  float v;
  if (c < 64)       v = x1[(size_t)b * 64 * N + (size_t)c * N + n];
  else if (c < 128) v = x2[(size_t)b * 64 * N + (size_t)(c - 64) * N + n];
  else              v = x3[(size_t)b * 64 * N + (size_t)(c - 128) * N + n];
  xc[(size_t)b * 192 * N + (size_t)c * N + n] = v;
}

// ---------------------------------------------------------------------------
// h = concat(x1,x2,x3, broadcast(gfeat), broadcast(lfeat)) -> (B,1280,N)
// ---------------------------------------------------------------------------
__global__ void build_h_kernel(const float* __restrict__ x1,
                               const float* __restrict__ x2,
                               const float* __restrict__ x3,
                               const float* __restrict__ gfeat,
                               const float* __restrict__ lfeat,
                               float* __restrict__ h) {
  const int N = NP;
  long i = (long)blockIdx.x * blockDim.x + threadIdx.x;
  long total = (long)BB * 1280 * N;
  if (i >= total) return;
  int n = (int)(i % N);
  long t = i / N;
  int c = (int)(t % 1280);
  int b = (int)(t / 1280);
  float v;
  if (c < 64)        v = x1[(size_t)b * 64 * N + (size_t)c * N + n];
  else if (c < 128)  v = x2[(size_t)b * 64 * N + (size_t)(c - 64) * N + n];
  else if (c < 192)  v = x3[(size_t)b * 64 * N + (size_t)(c - 128) * N + n];
  else if (c < 1216) v = gfeat[(size_t)b * 1024 + (c - 192)];
  else               v = lfeat[(size_t)b * 64 + (c - 1216)];
  h[(size_t)b * 1280 * N + (size_t)c * N + n] = v;
}

// ---------------------------------------------------------------------------
// Final conv P3 (6x128), input normalized+lrelu on the fly. out (B,6,N).
// ---------------------------------------------------------------------------
__global__ void final_kernel(const float* __restrict__ P3,
                             const float* __restrict__ Y,
                             const float* __restrict__ sc,
                             const float* __restrict__ sh,
                             float* __restrict__ out) {
  const int N = NP;
  int i = blockIdx.x * blockDim.x + threadIdx.x;
  if (i >= BB * N) return;
  int b = i / N, n = i % N;
  const float* yb = Y + (size_t)b * 128 * N + n;
  float acc[6] = {0.f, 0.f, 0.f, 0.f, 0.f, 0.f};
  for (int c = 0; c < 128; ++c) {
    float v = lrelu(yb[(size_t)c * N] * sc[c] + sh[c]);
#pragma unroll
    for (int m = 0; m < 6; ++m) acc[m] += P3[m * 128 + c] * v;
  }
#pragma unroll
  for (int m = 0; m < 6; ++m)
    out[(size_t)b * 6 * N + (size_t)m * N + n] = acc[m];
}

// ---------------------------------------------------------------------------
// Host orchestration
// ---------------------------------------------------------------------------
extern "C" void kernel_launch(void* const* d_in, const int* in_sizes, int n_in,
                              void* d_out, int out_size, void* d_ws,
                              size_t ws_size, hipStream_t stream) {
  (void)in_sizes; (void)n_in; (void)out_size; (void)ws_size;
  const float* points = (const float*)d_in[0];
  const float* label  = (const float*)d_in[1];
  const float* Wl[7]; const float* gl[7]; const float* bl[7];
  for (int i = 0; i < 7; ++i) {
    Wl[i] = (const float*)d_in[2 + 3 * i];
    gl[i] = (const float*)d_in[3 + 3 * i];
    bl[i] = (const float*)d_in[4 + 3 * i];
  }
  const float* Pm[4];
  for (int i = 0; i < 4; ++i) Pm[i] = (const float*)d_in[23 + i];
  const float* pg[3]; const float* pb[3];
  for (int i = 0; i < 3; ++i) {
    pg[i] = (const float*)d_in[27 + 2 * i];
    pb[i] = (const float*)d_in[28 + 2 * i];
  }
  float* out = (float*)d_out;
  float* ws  = (float*)d_ws;

  // Workspace arena (floats). Peak ~187 MB.
  float* yraw0 = ws + 0L;           // 12.8M  (B,64,N,K) raw conv out (ping)
  float* yraw1 = ws + 12800000L;    // 12.8M  (pong)
  float* ybig  = ws + 0L;           // 20.48M (B,1024,N) reuses edge buffers
  float* hbuf  = ws + 0L;           // 25.6M  (B,1280,N) reuses edge buffers
  float* yp0   = ws + 25600000L;    // 5.12M
  float* yp1   = ws + 30720000L;    // 5.12M
  float* yp2   = ws + 35840000L;    // 2.56M
  float* x1    = ws + 38400000L;    // 1.28M
  float* x2    = ws + 39680000L;
  float* x3    = ws + 40960000L;
  float* xc    = ws + 42240000L;    // 3.84M
  float* xxb   = ws + 46080000L;    // 20k
  int*   idx1  = (int*)(ws + 46100000L);
  int*   idx2  = (int*)(ws + 46300000L);
  int*   idx3  = (int*)(ws + 46500000L);
  float* stats = ws + 46700000L;    // 10 layers x 4096 (sum|sq|scale|shift)
  float* gfeat = ws + 46741000L;    // B*1024
  float* lfeat = ws + 46743100L;    // B*64

#define SUMp(L) (stats + (size_t)(L) * 4096)
#define SQp(L)  (stats + (size_t)(L) * 4096 + 1024)
#define SCp(L)  (stats + (size_t)(L) * 4096 + 2048)
#define SHp(L)  (stats + (size_t)(L) * 4096 + 3072)

  const int PKE = NP * KK;  // 100000 positions per batch for edge convs
  const float cntEdge = (float)(BB * PKE);
  const float cntPt   = (float)(BB * NP);
  dim3 gEdge((PKE + 127) / 128, 4, BB);   // M=64 edge-conv GEMMs
  dim3 gPt((NP + 127) / 128, 1, BB);
  int totBN = BB * NP;

  // ---------------- Stage 1: knn(points, C=3) -> conv0 -> conv1 -> max ----
  sqnorm_kernel<<<(totBN + 255) / 256, 256, 0, stream>>>(points, xxb, 3);
  knn_kernel<3><<<dim3(NP / 16, BB), 256, 0, stream>>>(points, xxb, idx1);

  hipMemsetAsync(SUMp(0), 0, 2048 * sizeof(float), stream);
  gemm_wmma_kernel<6, 3, true, false><<<gEdge, 128, 0, stream>>>(
      Wl[0], 64, nullptr, PKE, 0, nullptr, nullptr, points, idx1,
      yraw0, SUMp(0), SQp(0));
  finalize_bn_kernel<<<1, 64, 0, stream>>>(SUMp(0), SQp(0), gl[0], bl[0], 64,
                                           1.f / cntEdge, SCp(0), SHp(0));

  hipMemsetAsync(SUMp(1), 0, 2048 * sizeof(float), stream);
  gemm_wmma_kernel<64, 0, false, true><<<gEdge, 128, 0, stream>>>(
      Wl[1], 64, yraw0, PKE, (long)64 * PKE, SCp(0), SHp(0), nullptr, nullptr,
      yraw1, SUMp(1), SQp(1));
  finalize_bn_kernel<<<1, 64, 0, stream>>>(SUMp(1), SQp(1), gl[1], bl[1], 64,
                                           1.f / cntEdge, SCp(1), SHp(1));
  {
    long tot = (long)BB * 64 * NP;
    maxpool_k_kernel<<<(unsigned)((tot + 255) / 256), 256, 0, stream>>>(
        yraw1, SCp(1), SHp(1), x1, 64);
  }

  // ---------------- Stage 2: knn(x1, C=64) -> conv2 -> conv3 -> max -------
  sqnorm_kernel<<<(totBN + 255) / 256, 256, 0, stream>>>(x1, xxb, 64);
  knn_kernel<64><<<dim3(NP / 16, BB), 256, 0, stream>>>(x1, xxb, idx2);

  hipMemsetAsync(SUMp(2), 0, 2048 * sizeof(float), stream);
  gemm_wmma_kernel<128, 64, true, false><<<gEdge, 128, 0, stream>>>(
      Wl[2], 64, nullptr, PKE, 0, nullptr, nullptr, x1, idx2,
      yraw0, SUMp(2), SQp(2));
  finalize_bn_kernel<<<1, 64, 0, stream>>>(SUMp(2), SQp(2), gl[2], bl[2], 64,
                                           1.f / cntEdge, SCp(2), SHp(2));

  hipMemsetAsync(SUMp(3), 0, 2048 * sizeof(float), stream);
  gemm_wmma_kernel<64, 0, false, true><<<gEdge, 128, 0, stream>>>(
      Wl[3], 64, yraw0, PKE, (long)64 * PKE, SCp(2), SHp(2), nullptr, nullptr,
      yraw1, SUMp(3), SQp(3));
  finalize_bn_kernel<<<1, 64, 0, stream>>>(SUMp(3), SQp(3), gl[3], bl[3], 64,
                                           1.f / cntEdge, SCp(3), SHp(3));
  {
    long tot = (long)BB * 64 * NP;
    maxpool_k_kernel<<<(unsigned)((tot + 255) / 256), 256, 0, stream>>>(
        yraw1, SCp(3), SHp(3), x2, 64);
  }

  // ---------------- Stage 3: knn(x2, C=64) -> conv4 -> max ----------------
  sqnorm_kernel<<<(totBN + 255) / 256, 256, 0, stream>>>(x2, xxb, 64);
  knn_kernel<64><<<dim3(NP / 16, BB), 256, 0, stream>>>(x2, xxb, idx3);

  hipMemsetAsync(SUMp(4), 0, 2048 * sizeof(float), stream);
  gemm_wmma_kernel<128, 64, true, false><<<gEdge, 128, 0, stream>>>(
      Wl[4], 64, nullptr, PKE, 0, nullptr, nullptr, x2, idx3,
      yraw0, SUMp(4), SQp(4));
  finalize_bn_kernel<<<1, 64, 0, stream>>>(SUMp(4), SQp(4), gl[4], bl[4], 64,
                                           1.f / cntEdge, SCp(4), SHp(4));
  {
    long tot = (long)BB * 64 * NP;
    maxpool_k_kernel<<<(unsigned)((tot + 255) / 256), 256, 0, stream>>>(
        yraw0, SCp(4), SHp(4), x3, 64);
  }

  // ---------------- Global feature + label branch -------------------------
  {
    long tot = (long)BB * 192 * NP;
    concat3_kernel<<<(unsigned)((tot + 255) / 256), 256, 0, stream>>>(x1, x2,
                                                                      x3, xc);
  }
  hipMemsetAsync(SUMp(5), 0, 2048 * sizeof(float), stream);
  gemm_wmma_kernel<192, 0, false, false>
      <<<dim3((NP + 127) / 128, 64, BB), 128, 0, stream>>>(
          Wl[5], 1024, xc, NP, (long)192 * NP, nullptr, nullptr, nullptr,
          nullptr, ybig, SUMp(5), SQp(5));
  finalize_bn_kernel<<<4, 256, 0, stream>>>(SUMp(5), SQp(5), gl[5], bl[5],
                                            1024, 1.f / cntPt, SCp(5), SHp(5));
  gmax_kernel<<<dim3(1024, BB), 256, 0, stream>>>(ybig, SCp(5), SHp(5), gfeat,
                                                  1024);
  label_kernel<<<1, 64, 0, stream>>>(Wl[6], label, gl[6], bl[6], lfeat);

  {
    long tot = (long)BB * 1280 * NP;
    build_h_kernel<<<(unsigned)((tot + 255) / 256), 256, 0, stream>>>(
        x1, x2, x3, gfeat, lfeat, hbuf);
  }

  // ---------------- Point-wise MLP head -----------------------------------
  hipMemsetAsync(SUMp(6), 0, 2048 * sizeof(float), stream);
  gemm_wmma_kernel<1280, 0, false, false>
      <<<dim3((NP + 127) / 128, 16, BB), 128, 0, stream>>>(
          Pm[0], 256, hbuf, NP, (long)1280 * NP, nullptr, nullptr, nullptr,
          nullptr, yp0, SUMp(6), SQp(6));
  finalize_bn_kernel<<<1, 256, 0, stream>>>(SUMp(6), SQp(6), pg[0], pb[0], 256,
                                            1.f / cntPt, SCp(6), SHp(6));

  hipMemsetAsync(SUMp(7), 0, 2048 * sizeof(float), stream);
  gemm_wmma_kernel<256, 0, false, true>
      <<<dim3((NP + 127) / 128, 16, BB), 128, 0, stream>>>(
          Pm[1], 256, yp0, NP, (long)256 * NP, SCp(6), SHp(6), nullptr,
          nullptr, yp1, SUMp(7), SQp(7));
  finalize_bn_kernel<<<1, 256, 0, stream>>>(SUMp(7), SQp(7), pg[1], pb[1], 256,
                                            1.f / cntPt, SCp(7), SHp(7));

  hipMemsetAsync(SUMp(8), 0, 2048 * sizeof(float), stream);
  gemm_wmma_kernel<256, 0, false, true>
      <<<dim3((NP + 127) / 128, 8, BB), 128, 0, stream>>>(
          Pm[2], 128, yp1, NP, (long)256 * NP, SCp(7), SHp(7), nullptr,
          nullptr, yp2, SUMp(8), SQp(8));
  finalize_bn_kernel<<<1, 128, 0, stream>>>(SUMp(8), SQp(8), pg[2], pb[2], 128,
                                            1.f / cntPt, SCp(8), SHp(8));

  final_kernel<<<(BB * NP + 255) / 256, 256, 0, stream>>>(Pm[3], yp2, SCp(8),
                                                          SHp(8), out);
#undef SUMp
#undef SQp
#undef SCp
#undef SHp
}